// GraphSpatialLoss3D_19061064859676
// MI455X (gfx1250) — compile-verified
//
#include <hip/hip_runtime.h>
#include <hip/hip_bf16.h>

// GraphSpatialLoss3D for MI455X (gfx1250).
// Bandwidth-bound streaming reduction: 134 MB @ 23.3 TB/s => ~5.8 us ideal.
// CDNA5 async global->LDS pipeline (ASYNCcnt): each wave keeps a 4-deep queue
// of 512B transfers in flight with LDS (320KB/WGP) as the landing zone.
// Index structure exploited: block <-> one y-slice (y == blk), wave-round <->
// one z-row (x == 8r+wave, z0 == 4*lane), so per-element VALU is ~3 ops and
// the y-moment collapses to blk*s after reduction. WMMA intentionally unused:
// a weighted-sum reduction cannot utilize a 16x16 WMMA tile; HBM is the limit.

#define TPB     256
#define ROUNDS  16
#define DEPTH   4
#define BPC     128                    // blocks per channel == H (one y per block)
#define CH_F4   524288                 // 128^3 / 4 float4 per channel
#define NCH     16                     // B*C = 2*8

#if defined(__HIP_DEVICE_COMPILE__)

#define LDS_AS __attribute__((address_space(3)))
#define GLB_AS __attribute__((address_space(1)))

// Matches the builtin's parameter type exactly.
typedef int b128_t __attribute__((vector_size(16)));

#if __has_builtin(__builtin_amdgcn_global_load_async_to_lds_b128)
#define ASYNC_CP16(gsrc, ldst)                                                 \
  __builtin_amdgcn_global_load_async_to_lds_b128(                              \
      (GLB_AS b128_t*)(gsrc), (LDS_AS b128_t*)(ldst), 0, 0)
#else
#define ASYNC_CP16(gsrc, ldst)                                                 \
  asm volatile("global_load_async_to_lds_b128 %0, %1, off"                     \
               :: "v"((unsigned)(unsigned long long)(LDS_AS void*)(ldst)),     \
                  "v"((GLB_AS void*)(gsrc))                                    \
               : "memory")
#endif

#if __has_builtin(__builtin_amdgcn_s_wait_asynccnt)
#define WAIT_ASYNC(n)                                                          \
  do { __builtin_amdgcn_s_wait_asynccnt(n); asm volatile("" ::: "memory"); } while (0)
#else
#define WAIT_ASYNC(n) asm volatile("s_wait_asynccnt %0" :: "i"(n) : "memory")
#endif

#else  // host pass: bodies are parsed but not codegen'd; keep them trivial
#define ASYNC_CP16(gsrc, ldst) ((void)0)
#define WAIT_ASYNC(n)          ((void)0)
#endif

// ---------------------------------------------------------------------------
// Kernel 1: per-block partial sums (s, s*x, t*z) for one y-slice of a channel.
// blockIdx.x = chan*BPC + blk, blk == y. Each wave consumes only its own LDS
// slice, so the async pipeline needs no __syncthreads -- only per-wave
// ASYNCcnt waits.
// ---------------------------------------------------------------------------
__global__ __launch_bounds__(TPB) void gsl3d_partials(
    const float4* __restrict__ in, float* __restrict__ ws) {
  const int tid  = threadIdx.x;
  const int chan = blockIdx.x / BPC;
  const int blk  = blockIdx.x % BPC;            // == y index of this slice
  const int wave = tid >> 5, lane = tid & 31;
  const float4* gbase =
      in + (size_t)chan * CH_F4 + (size_t)blk * (ROUNDS * TPB);

  __shared__ float4 buf[DEPTH][TPB];

  // Prime the pipeline DEPTH-1 deep.
  ASYNC_CP16(gbase + 0 * TPB + tid, &buf[0][tid]);
  ASYNC_CP16(gbase + 1 * TPB + tid, &buf[1][tid]);
  ASYNC_CP16(gbase + 2 * TPB + tid, &buf[2][tid]);

  const float zf = (float)(lane << 2);          // z0 of this lane's float4
  const float xw = (float)wave;                 // x = 8r + wave (exact, no wrap)

  float s = 0.f, sx = 0.f, sz = 0.f;

#pragma unroll
  for (int r = 0; r < ROUNDS; ++r) {
    if (r + (DEPTH - 1) < ROUNDS) {
      ASYNC_CP16(gbase + (r + DEPTH - 1) * TPB + tid,
                 &buf[(r + DEPTH - 1) % DEPTH][tid]);
      WAIT_ASYNC(DEPTH - 1);     // in-order completion => round r has landed
    } else if (r == ROUNDS - 3) { WAIT_ASYNC(2); }
    else if (r == ROUNDS - 2)   { WAIT_ASYNC(1); }
    else                        { WAIT_ASYNC(0); }

    const float4 v = buf[r % DEPTH][tid];

    const float t0 = v.x > 0.5f ? v.x : 0.0f;
    const float t1 = v.y > 0.5f ? v.y : 0.0f;
    const float t2 = v.z > 0.5f ? v.z : 0.0f;
    const float t3 = v.w > 0.5f ? v.w : 0.0f;
    const float ls = (t0 + t1) + (t2 + t3);
    // z-moment: t0*z0 + t1*(z0+1) + t2*(z0+2) + t3*(z0+3)
    //         = z0*ls + (t1 + 2*t2 + 3*t3)
    const float w3 = __builtin_fmaf(3.0f, t3, __builtin_fmaf(2.0f, t2, t1));
    s  += ls;
    sx += ls * (xw + (float)(8 * r));           // x weight, raw index
    sz += __builtin_fmaf(zf, ls, w3);           // z weight, raw index
  }

  // wave32 butterfly reduction (3 accumulators; y-moment is blk*s)
#pragma unroll
  for (int off = 16; off > 0; off >>= 1) {
    s  += __shfl_xor(s,  off, 32);
    sx += __shfl_xor(sx, off, 32);
    sz += __shfl_xor(sz, off, 32);
  }

  __shared__ float part[TPB / 32][3];
  if (lane == 0) {
    part[wave][0] = s; part[wave][1] = sx; part[wave][2] = sz;
  }
  __syncthreads();
  if (tid == 0) {
    float S = 0.f, SX = 0.f, SZ = 0.f;
#pragma unroll
    for (int w = 0; w < TPB / 32; ++w) {
      S += part[w][0]; SX += part[w][1]; SZ += part[w][2];
    }
    float* o = ws + (size_t)blockIdx.x * 4;
    o[0] = S;                  // mass
    o[1] = S * (float)blk;     // y-moment: y is constant over the slice
    o[2] = SX;                 // x-moment (raw index)
    o[3] = SZ;                 // z-moment (raw index)
  }
}

// ---------------------------------------------------------------------------
// Kernel 2: deterministic final reduction + 8-relation loss (single block).
// ---------------------------------------------------------------------------
__global__ __launch_bounds__(64) void gsl3d_finalize(
    const float* __restrict__ ws, float* __restrict__ out) {
  __shared__ float sums[NCH][4];
  const int t = threadIdx.x;
  {
    const int ch = t >> 2, comp = t & 3;
    float a = 0.f;
#pragma unroll 16
    for (int i = 0; i < BPC; ++i) a += ws[(size_t)(ch * BPC + i) * 4 + comp];
    sums[ch][comp] = a;
  }
  __syncthreads();

  if (t == 0) {
    float cy[NCH], cx[NCH], cz[NCH];
#pragma unroll
    for (int ch = 0; ch < NCH; ++ch) {
      const float sden = 128.0f * sums[ch][0];   // /H (=W=D) normalization
      cy[ch] = sums[ch][1] / sden;               // 0/0 -> NaN, matches ref
      cx[ch] = sums[ch][2] / sden;
      cz[ch] = sums[ch][3] / sden;
    }
    const int   ri[8] = {0, 1, 2, 3, 4, 5, 6, 0};
    const int   rj[8] = {1, 2, 3, 4, 5, 6, 7, 7};
    const float gy[8] = {0.1f, 0.0f, -0.1f, 0.0f, 0.05f, 0.0f, 0.1f, -0.05f};
    const float gx[8] = {0.0f, 0.1f, 0.05f, 0.0f, -0.05f, 0.1f, 0.0f, 0.05f};
    const float gz[8] = {0.05f, 0.0f, 0.0f, 0.1f, 0.0f, -0.1f, 0.0f, 0.05f};

    float loss = 0.f;
#pragma unroll
    for (int b = 0; b < 2; ++b) {
#pragma unroll
      for (int r = 0; r < 8; ++r) {
        const int i = b * 8 + ri[r], j = b * 8 + rj[r];
        float dy = cy[i] - cy[j] - gy[r];
        float dx = cx[i] - cx[j] - gx[r];
        float dz = cz[i] - cz[j] - gz[r];
        if (!__builtin_isfinite(dy)) dy = 0.f;   // nan_to_num(nan/inf -> 0)
        if (!__builtin_isfinite(dx)) dx = 0.f;
        if (!__builtin_isfinite(dz)) dz = 0.f;
        loss += 0.5f * (dy * dy + dx * dx + dz * dz);  // mean over batch B=2
      }
    }
    out[0] = loss;
  }
}

extern "C" void kernel_launch(void* const* d_in, const int* in_sizes, int n_in,
                              void* d_out, int out_size, void* d_ws, size_t ws_size,
                              hipStream_t stream) {
  (void)in_sizes; (void)n_in; (void)out_size; (void)ws_size;
  const float4* in = (const float4*)d_in[0];
  float* ws  = (float*)d_ws;      // 16*128*4 floats = 32 KB, fully overwritten
  float* out = (float*)d_out;

  gsl3d_partials<<<dim3(NCH * BPC), dim3(TPB), 0, stream>>>(in, ws);
  gsl3d_finalize<<<dim3(1), dim3(64), 0, stream>>>(ws, out);
}